// RQSLayer_5540507812121
// MI455X (gfx1250) — compile-verified
//
#include <hip/hip_runtime.h>
#include <cmath>

// ---------------------------------------------------------------------------
// RQS (rational-quadratic spline) forward, N=2M rows, 25 params/row.
// Memory-bound: ~224MB traffic -> ~9.6us floor @ 23.3TB/s. No GEMM -> no WMMA.
// Strategy: coalesced b128 tile staging into LDS via gfx1250 async-to-LDS
// (ASYNCcnt double buffering), conflict-free LDS row reads (stride 25 dwords,
// coprime with 64 banks), index-first gather so only 2 softplus evals/row,
// non-temporal output stores.
// ---------------------------------------------------------------------------

#define BINS 8
static constexpr float MIN_W = 0.001f;
static constexpr float MIN_H = 0.001f;
static constexpr float MIN_D = 0.001f;
static constexpr float TAILB = 3.0f;

static constexpr int TPB           = 256;
static constexpr int ROWS_PER_TILE = 256;
static constexpr int TH_PER_ROW    = 3 * BINS + 1;                  // 25 floats
static constexpr int TILE_F        = ROWS_PER_TILE * TH_PER_ROW;    // 6400 floats
static constexpr int TILE_CHUNKS   = TILE_F / 4;                    // 1600 b128 chunks
static constexpr int CHUNK_ITERS   = (TILE_CHUNKS + TPB - 1) / TPB; // 7

typedef float v4f __attribute__((ext_vector_type(4)));
typedef int   v4i __attribute__((ext_vector_type(4)));
typedef __attribute__((address_space(1))) v4i glb_v4i;  // global (AS1)
typedef __attribute__((address_space(3))) v4i lds_v4i;  // LDS (AS3, 32-bit ptr)

#if defined(__AMDGCN__) && __has_builtin(__builtin_amdgcn_global_load_async_to_lds_b128)
#define USE_ASYNC_LDS 1
#else
#define USE_ASYNC_LDS 0
#endif

// Stage `nchunk` b128 chunks starting at theta[baseF] into LDS at sdst.
// Uniform instruction count per wave (index clamp, no divergent branch) so
// the per-wave ASYNCcnt bookkeeping below is exact.
__device__ __forceinline__ void stage_tile(const float* __restrict__ theta,
                                           long long baseF, int nchunk,
                                           float* sdst) {
#pragma unroll
  for (int j = 0; j < CHUNK_ITERS; ++j) {
    int k = j * TPB + (int)threadIdx.x;
    k = (k < nchunk) ? k : (nchunk - 1);  // clamp: redundant dup copies are benign
#if USE_ASYNC_LDS
    // AS-qualified typed pointers via integer round-trip (always legal in C++).
    glb_v4i* gp = (glb_v4i*)(unsigned long long)(uintptr_t)(theta + baseF + (long long)k * 4);
    lds_v4i* lp = (lds_v4i*)(unsigned)(uintptr_t)(sdst + k * 4);  // low 32 bits = LDS offset
    __builtin_amdgcn_global_load_async_to_lds_b128(gp, lp, /*offset=*/0, /*cpol=*/0);
#else
    const v4f* gp = (const v4f*)(theta + baseF) + k;
    v4f v = __builtin_nontemporal_load(gp);
    *(v4f*)(sdst + k * 4) = v;
#endif
  }
}

template <int NWAIT>
__device__ __forceinline__ void wait_async() {
#if USE_ASYNC_LDS
#if __has_builtin(__builtin_amdgcn_s_wait_asynccnt)
  __builtin_amdgcn_s_wait_asynccnt(NWAIT);
#else
  asm volatile("s_wait_asynccnt %0" ::"n"(NWAIT) : "memory");
#endif
#endif
}

__global__ __launch_bounds__(TPB) void rqs_fwd_kernel(
    const float* __restrict__ theta, const float* __restrict__ y,
    float* __restrict__ out, int N) {
  __shared__ __align__(16) float sbuf[2][TILE_F];

  const int ntiles = (N + ROWS_PER_TILE - 1) / ROWS_PER_TILE;
  int t = (int)blockIdx.x;
  if (t >= ntiles) return;

  auto nchunks_of = [&](int tile) {
    int rows = N - tile * ROWS_PER_TILE;
    rows = rows > ROWS_PER_TILE ? ROWS_PER_TILE : rows;
    return (rows * TH_PER_ROW) >> 2;  // rows*25 is a multiple of 4 for this N
  };

  int p = 0;
  stage_tile(theta, (long long)t * TILE_F, nchunks_of(t), sbuf[0]);

  for (; t < ntiles; t += (int)gridDim.x) {
    const int tn = t + (int)gridDim.x;
    const bool havenext = tn < ntiles;
    if (havenext)
      stage_tile(theta, (long long)tn * TILE_F, nchunks_of(tn), sbuf[p ^ 1]);
    // In-order ASYNCcnt retirement: <=CHUNK_ITERS outstanding => current tile landed.
    if (havenext) wait_async<CHUNK_ITERS>(); else wait_async<0>();
    __syncthreads();

    const int row = t * ROWS_PER_TILE + (int)threadIdx.x;
    const float* th = &sbuf[p][threadIdx.x * TH_PER_ROW];  // stride 25: bank-conflict-free

    if (row < N) {
      const float Bt = TAILB;
      const float x = y[row];

      float e[BINS];
      // --- widths: softmax -> knots cw[0..8] ---
      float m = th[0];
#pragma unroll
      for (int i = 1; i < BINS; ++i) m = fmaxf(m, th[i]);
      float s = 0.f;
#pragma unroll
      for (int i = 0; i < BINS; ++i) { e[i] = __expf(th[i] - m); s += e[i]; }
      float cw[BINS + 1];
      cw[0] = -Bt;
      {
        const float scale = (1.0f - MIN_W * (float)BINS) / s;
        float acc = 0.f;
#pragma unroll
        for (int i = 0; i < BINS; ++i) {
          acc += MIN_W + scale * e[i];
          cw[i + 1] = 2.f * Bt * acc - Bt;
        }
      }
      cw[BINS] = Bt;

      // --- heights: softmax -> knots ch[0..8] ---
      m = th[BINS];
#pragma unroll
      for (int i = 1; i < BINS; ++i) m = fmaxf(m, th[BINS + i]);
      s = 0.f;
#pragma unroll
      for (int i = 0; i < BINS; ++i) { e[i] = __expf(th[BINS + i] - m); s += e[i]; }
      float ch[BINS + 1];
      ch[0] = -Bt;
      {
        const float scale = (1.0f - MIN_H * (float)BINS) / s;
        float acc = 0.f;
#pragma unroll
        for (int i = 0; i < BINS; ++i) {
          acc += MIN_H + scale * e[i];
          ch[i + 1] = 2.f * Bt * acc - Bt;
        }
      }
      ch[BINS] = Bt;

      // --- bin search (select-based, branch-free) ---
      const float xc = fminf(fmaxf(x, -Bt), Bt);
      int idx = 0;
#pragma unroll
      for (int j = 1; j < BINS; ++j) idx += (xc >= cw[j]) ? 1 : 0;

      // --- gather knots + the two derivative logits we actually need ---
      float in_cw = cw[0], in_cw1 = cw[1];
      float in_ch = ch[0], in_ch1 = ch[1];
      float u0 = th[2 * BINS], u1 = th[2 * BINS + 1];
#pragma unroll
      for (int j = 1; j < BINS; ++j) {
        const bool c = (idx >= j);
        in_cw  = c ? cw[j]     : in_cw;
        in_cw1 = c ? cw[j + 1] : in_cw1;
        in_ch  = c ? ch[j]     : in_ch;
        in_ch1 = c ? ch[j + 1] : in_ch1;
        u0 = c ? th[2 * BINS + j]     : u0;
        u1 = c ? th[2 * BINS + j + 1] : u1;
      }

      // only 2 softplus evals/row (reference does 9; only gathered ones matter)
      const float d0 = MIN_D + fmaxf(u0, 0.f) + log1pf(__expf(-fabsf(u0)));
      const float d1 = MIN_D + fmaxf(u1, 0.f) + log1pf(__expf(-fabsf(u1)));

      const float w_   = in_cw1 - in_cw;
      const float h_   = in_ch1 - in_ch;
      const float invw = 1.f / w_;
      const float tt   = (xc - in_cw) * invw;
      const float dlt  = h_ * invw;
      const float omt  = 1.f - tt;
      const float t1m  = tt * omt;
      const float t2   = tt * tt;
      const float num  = h_ * (dlt * t2 + d0 * t1m);
      const float den  = dlt + (d0 + d1 - 2.f * dlt) * t1m;
      float outv = in_ch + num / den;
      const float dnum = dlt * dlt * (d1 * t2 + 2.f * dlt * t1m + d0 * omt * omt);
      float ldv = __logf(dnum) - 2.f * __logf(den);

      const bool inside = (x >= -Bt) && (x <= Bt);
      outv = inside ? outv : x;
      ldv  = inside ? ldv : 0.f;

      __builtin_nontemporal_store(outv, &out[row]);      // out
      __builtin_nontemporal_store(ldv, &out[N + row]);   // logdet
    }
    __syncthreads();  // all reads of sbuf[p] done before it is refilled
    p ^= 1;
  }
}

extern "C" void kernel_launch(void* const* d_in, const int* in_sizes, int n_in,
                              void* d_out, int out_size, void* d_ws, size_t ws_size,
                              hipStream_t stream) {
  const float* theta = (const float*)d_in[0];  // [N, 25] f32
  const float* y     = (const float*)d_in[1];  // [N, 1]  f32
  float* out = (float*)d_out;                  // [N] out then [N] logdet
  const int N = in_sizes[1];

  const int ntiles = (N + ROWS_PER_TILE - 1) / ROWS_PER_TILE;
  int nblk = ntiles < 3072 ? ntiles : 3072;    // persistent blocks; >=2 tiles each
  if (nblk < 1) nblk = 1;
  hipLaunchKernelGGL(rqs_fwd_kernel, dim3(nblk), dim3(TPB), 0, stream,
                     theta, y, out, N);
}